// LaughingHyenaModalFilter_481036337251
// MI455X (gfx1250) — compile-verified
//
#include <hip/hip_runtime.h>

#define TCHUNK 16
#define KMAX   64
#define BLOCK  128

typedef float v4f __attribute__((ext_vector_type(4)));

template <int KC>
__global__ __launch_bounds__(BLOCK)
void LaughingHyenaModalFilter_kernel(const float* __restrict__ rr,
                                     const float* __restrict__ th,
                                     const float* __restrict__ Rre,
                                     const float* __restrict__ Rim,
                                     const float* __restrict__ h0,
                                     float* __restrict__ out,
                                     int Krt, int D, int L) {
    const int K = (KC > 0) ? KC : Krt;   // compile-time K when specialized
    __shared__ float smem[4 * KMAX];

    const int d   = blockIdx.x;
    const int tid = threadIdx.x;

    // ---- Stage r/theta/R_re/R_im columns for channel d into LDS via CDNA5
    // ---- async global->LDS DMA (ASYNCcnt-tracked). One b32 per (array,k).
    const float* bases[4] = { rr, th, Rre, Rim };
#pragma unroll
    for (int a = 0; a < 4; ++a) {
        const float* bp = bases[a];
        for (int k = tid; k < K; k += BLOCK) {
            unsigned long long ga =
                (unsigned long long)(const void*)(bp + (size_t)k * (size_t)D + (size_t)d);
            // low 32 bits of a generic pointer to LDS == LDS byte offset
            unsigned lo = (unsigned)(size_t)(const void*)&smem[a * KMAX + k];
            asm volatile("global_load_async_to_lds_b32 %0, %1, off"
                         :: "v"(lo), "v"(ga) : "memory");
        }
    }
#if defined(__has_builtin) && __has_builtin(__builtin_amdgcn_s_wait_asynccnt)
    __builtin_amdgcn_s_wait_asynccnt(0);
#else
    asm volatile("s_wait_asynccnt 0" ::: "memory");
#endif
    __syncthreads();   // split barrier: s_barrier_signal -1 / s_barrier_wait -1

    // ---- Each thread produces TCHUNK consecutive output samples for channel d.
    const int nchunk = L / TCHUNK;
    const int chunk  = blockIdx.y * BLOCK + tid;
    if (chunk >= nchunk) return;

    const int   tau0 = chunk * TCHUNK;
    // output row d: [h0, g(0), g(1), ..., g(L-2)] where g(e) = sum_k Re(R p^e)
    const float e0 = (chunk == 0) ? 0.0f : (float)(tau0 - 1);

    float acc[TCHUNK];
#pragma unroll
    for (int j = 0; j < TCHUNK; ++j) acc[j] = 0.0f;

#pragma unroll 2   // two independent complex-recurrence chains in flight
    for (int k = 0; k < K; ++k) {
        const float rk  = smem[k];
        const float tk  = smem[KMAX + k];
        const float cre = smem[2 * KMAX + k];
        const float cim = smem[3 * KMAX + k];

        // p = r * e^{i*theta}
        float ps, pc;
        __sincosf(tk, &ps, &pc);
        const float pr = rk * pc;
        const float pi = rk * ps;

        // s = R * p^{e0}  (polar jump to chunk start)
        const float mag = __expf(e0 * __logf(rk));
        float sa, ca;
        __sincosf(e0 * tk, &sa, &ca);
        float sr = mag * (cre * ca - cim * sa);
        float si = mag * (cre * sa + cim * ca);

        // 16-step complex recurrence: s <- s * p, accumulate Re(s)
#pragma unroll
        for (int j = 0; j < TCHUNK; ++j) {
            acc[j] += sr;
            const float nr = sr * pr - si * pi;
            const float ni = sr * pi + si * pr;
            sr = nr;
            si = ni;
        }
    }

    float fin[TCHUNK];
    if (chunk == 0) {
        fin[0] = h0[d];
#pragma unroll
        for (int j = 1; j < TCHUNK; ++j) fin[j] = acc[j - 1];
    } else {
#pragma unroll
        for (int j = 0; j < TCHUNK; ++j) fin[j] = acc[j];
    }

    // ---- Write-once output: non-temporal 128-bit stores (th:NT), 64B-aligned.
    float* op = out + (size_t)d * (size_t)L + (size_t)tau0;
#pragma unroll
    for (int v = 0; v < TCHUNK / 4; ++v) {
        v4f x = { fin[4 * v + 0], fin[4 * v + 1], fin[4 * v + 2], fin[4 * v + 3] };
        __builtin_nontemporal_store(x, (v4f*)(op + 4 * v));
    }
}

extern "C" void kernel_launch(void* const* d_in, const int* in_sizes, int n_in,
                              void* d_out, int out_size, void* d_ws, size_t ws_size,
                              hipStream_t stream) {
    const float* rr  = (const float*)d_in[0];
    const float* th  = (const float*)d_in[1];
    const float* Rre = (const float*)d_in[2];
    const float* Rim = (const float*)d_in[3];
    const float* h0  = (const float*)d_in[4];

    const int D = in_sizes[4];            // h_0 has D elements
    const int K = in_sizes[0] / D;        // r is (K, D)
    const int L = out_size / D;           // output is (D, L)

    const int nchunk = L / TCHUNK;        // L is a multiple of 16 (2048)
    dim3 grid(D, (nchunk + BLOCK - 1) / BLOCK);

    if (K == 32) {
        LaughingHyenaModalFilter_kernel<32><<<grid, BLOCK, 0, stream>>>(
            rr, th, Rre, Rim, h0, (float*)d_out, K, D, L);
    } else {
        LaughingHyenaModalFilter_kernel<0><<<grid, BLOCK, 0, stream>>>(
            rr, th, Rre, Rim, h0, (float*)d_out, K, D, L);
    }
}